// MessagePassingLayerEC_87110526697697
// MI455X (gfx1250) — compile-verified
//
#include <hip/hip_runtime.h>

#define DIM 128

typedef float v2f __attribute__((ext_vector_type(2)));
typedef float v8f __attribute__((ext_vector_type(8)));

// ---------------------------------------------------------------------------
// Projection GEMM: out[m][n] = sum_k node[m][k] * W[k][n] + b[n]
// One wave computes one 16x16 output tile using V_WMMA_F32_16X16X4_F32.
// Block = 256 threads = 8 waves -> one 16-row x 128-col strip per block.
// blockIdx.y selects (W_src,b_src)->src_proj vs (W_dest,b_dest)->dest_proj.
// ---------------------------------------------------------------------------
__global__ void __launch_bounds__(256)
proj_gemm_kernel(const float* __restrict__ node,
                 const float* __restrict__ Wsrc, const float* __restrict__ bsrc,
                 const float* __restrict__ Wdst, const float* __restrict__ bdst,
                 float* __restrict__ src_proj, float* __restrict__ dest_proj,
                 int V)
{
    const int lane    = threadIdx.x & 31;
    const int wave    = threadIdx.x >> 5;
    const int m0      = blockIdx.x * 16;
    const int n0      = wave * 16;
    const int halfsel = lane >> 4;     // 0 for lanes 0-15, 1 for lanes 16-31
    const int l15     = lane & 15;

    const bool isDst  = (blockIdx.y != 0);
    const float* __restrict__ W   = isDst ? Wdst : Wsrc;
    const float* __restrict__ b   = isDst ? bdst : bsrc;
    float* __restrict__       out = isDst ? dest_proj : src_proj;

    const int arow = m0 + l15;   // A-matrix row this lane supplies
    const int bcol = n0 + l15;   // B-matrix column this lane supplies

    // A: 16x4 f32 tile. Lane layout (ISA 7.12.2, 32-bit A 16x4):
    //   VGPR0 = A[arow][k + 2*halfsel], VGPR1 = A[arow][k + 2*halfsel + 1]
    // -> contiguous float2 load per K-step.
    const float* __restrict__ aptr = node + (size_t)arow * DIM + 2 * halfsel;
    // B: 4x16 f32 tile. By C/D-style row striping:
    //   VGPR0 = W[k + 2*halfsel][bcol], VGPR1 = W[k + 2*halfsel + 1][bcol]
    const float* __restrict__ bptr = W + (size_t)(2 * halfsel) * DIM + bcol;

    v8f acc = {};
#pragma unroll
    for (int k = 0; k < DIM; k += 4) {
        v2f a = *(const v2f*)(aptr + k);          // 8B-aligned contiguous
        v2f bb;
        bb.x = bptr[(size_t)(k + 0) * DIM];
        bb.y = bptr[(size_t)(k + 1) * DIM];
        // D = A(16x4) x B(4x16) + C ; 8-arg form:
        // (neg_a, A, neg_b, B, c_mod, C, reuse_a, reuse_b)
        acc = __builtin_amdgcn_wmma_f32_16x16x4_f32(
                  false, a, false, bb, (short)0, acc, false, false);
    }

    // C/D layout: VGPR r -> row m0 + r + 8*halfsel, col bcol.
    const float bias = b[bcol];
#pragma unroll
    for (int r = 0; r < 8; ++r) {
        const int m = m0 + r + 8 * halfsel;
        out[(size_t)m * DIM + bcol] = acc[r] + bias;
    }
}

// ---------------------------------------------------------------------------
// Edge phase: one wave per edge, each lane owns 4 contiguous f32 (128/32).
//   msg = relu(src_proj[s] + dest_proj[d] + edge_emb[c])
//   out[d] += msg   (non-returning global_atomic_add_f32, L2-resident)
// ---------------------------------------------------------------------------
__global__ void __launch_bounds__(256)
edge_scatter_kernel(const float* __restrict__ src_proj,
                    const float* __restrict__ dest_proj,
                    const int* __restrict__ edge_src,
                    const int* __restrict__ edge_dest,
                    const int* __restrict__ edge_cls,
                    const float* __restrict__ edge_emb,
                    float* __restrict__ out, int E)
{
    const int e = blockIdx.x * 8 + (threadIdx.x >> 5);   // 8 edges per block
    if (e >= E) return;
    const int lane = threadIdx.x & 31;
    const int off  = lane * 4;

    const int s = edge_src[e];
    const int d = edge_dest[e];
    const int c = edge_cls[e];

    const float4 sp = *(const float4*)(src_proj  + (size_t)s * DIM + off);
    const float4 dp = *(const float4*)(dest_proj + (size_t)d * DIM + off);
    const float4 em = *(const float4*)(edge_emb  + (size_t)c * DIM + off);

    float4 m;
    m.x = fmaxf(sp.x + dp.x + em.x, 0.0f);
    m.y = fmaxf(sp.y + dp.y + em.y, 0.0f);
    m.z = fmaxf(sp.z + dp.z + em.z, 0.0f);
    m.w = fmaxf(sp.w + dp.w + em.w, 0.0f);

    float* o = out + (size_t)d * DIM + off;
    unsafeAtomicAdd(o + 0, m.x);
    unsafeAtomicAdd(o + 1, m.y);
    unsafeAtomicAdd(o + 2, m.z);
    unsafeAtomicAdd(o + 3, m.w);
}

extern "C" void kernel_launch(void* const* d_in, const int* in_sizes, int n_in,
                              void* d_out, int out_size, void* d_ws, size_t ws_size,
                              hipStream_t stream) {
    const float* node = (const float*)d_in[0];
    const int*   esrc = (const int*)d_in[1];
    const int*   edst = (const int*)d_in[2];
    const int*   ecls = (const int*)d_in[3];
    const float* Wsrc = (const float*)d_in[4];
    const float* bsrc = (const float*)d_in[5];
    const float* Wdst = (const float*)d_in[6];
    const float* bdst = (const float*)d_in[7];
    const float* eemb = (const float*)d_in[8];
    float* out = (float*)d_out;

    const int V = in_sizes[0] / DIM;   // 50000
    const int E = in_sizes[1];         // 640000

    float* src_proj  = (float*)d_ws;
    float* dest_proj = src_proj + (size_t)V * DIM;

    // Output accumulates via atomics; zero it every call (harness poisons it).
    hipMemsetAsync(d_out, 0, (size_t)V * DIM * sizeof(float), stream);

    // GEMM: V/16 row-tiles, y=0 -> src projection, y=1 -> dest projection.
    dim3 ggrid((V + 15) / 16, 2);
    proj_gemm_kernel<<<ggrid, dim3(256), 0, stream>>>(
        node, Wsrc, bsrc, Wdst, bdst, src_proj, dest_proj, V);

    // Edge gather + ReLU + atomic scatter-add.
    const int nblocks = (E + 7) / 8;
    edge_scatter_kernel<<<nblocks, dim3(256), 0, stream>>>(
        src_proj, dest_proj, esrc, edst, ecls, eemb, out, E);
}